// UpsampleLayer_27127013441598
// MI455X (gfx1250) — compile-verified
//
#include <hip/hip_runtime.h>
#include <cstdint>
#include <cstddef>

#define BB   8
#define NN   2048
#define CDIM 128
#define HIDD 256
#define KNNK 16
#define UPR  4
#define NHYP 43
#define OUTW (NHYP*UPR)   /* 172 */
#define OUTS (1*UPR)      /* 4   */
#define OUTF (CDIM*UPR)   /* 512 */
#define HPITCH 260        /* LDS pitch for H tile: 260 % 64 == 4 -> conflict-free */

typedef float v2f __attribute__((ext_vector_type(2)));
typedef float v8f __attribute__((ext_vector_type(8)));

static __device__ __forceinline__ v8f wmma_f32(v2f a, v2f b, v8f c) {
  // D = A(16x4 f32) * B(4x16 f32) + C(16x16 f32)
  return __builtin_amdgcn_wmma_f32_16x16x4_f32(false, a, false, b, (short)0, c,
                                               false, false);
}

// ---------------------------------------------------------------------------
// 1) squared feature norms  sq[b,n] = sum_c feats[b,c,n]^2
// ---------------------------------------------------------------------------
__global__ __launch_bounds__(256) void k_sq(const float* __restrict__ feats,
                                            float* __restrict__ sq) {
  int t = blockIdx.x * blockDim.x + threadIdx.x;
  if (t >= BB * NN) return;
  int b = t / NN, n = t % NN;
  const float* f = feats + (size_t)b * CDIM * NN + n;
  float s = 0.f;
  #pragma unroll 8
  for (int c = 0; c < CDIM; ++c) { float v = f[(size_t)c * NN]; s += v * v; }
  sq[t] = s;
}

// ---------------------------------------------------------------------------
// 2) feature-space KNN: per row m, top-16 of  2*dot - sq[m] - sq[n]
//    one wave per row; 8 rows per workgroup; per-lane sorted top-16 then
//    wave-wide argmax merge (16 rounds of xor-butterfly reduction).
// ---------------------------------------------------------------------------
__global__ __launch_bounds__(256) void k_topk(const float* __restrict__ feats,
                                              const float* __restrict__ sq,
                                              int* __restrict__ idxo) {
  __shared__ float arow[8 * CDIM];
  int b  = blockIdx.x / (NN / 8);
  int m0 = (blockIdx.x % (NN / 8)) * 8;
  int w = threadIdx.x >> 5, lane = threadIdx.x & 31;
  int m = m0 + w;
  const float* fb = feats + (size_t)b * CDIM * NN;

  #pragma unroll
  for (int i = 0; i < CDIM / 32; ++i) {
    int c = lane + 32 * i;
    arow[w * CDIM + c] = fb[(size_t)c * NN + m];
  }
  __syncthreads();

  float sqm = sq[b * NN + m];
  float tv[KNNK]; int ti[KNNK];
  #pragma unroll
  for (int j = 0; j < KNNK; ++j) { tv[j] = -3.402823466e38f; ti[j] = 0x7FFFFFFF; }

  for (int it = 0; it < NN / 32; ++it) {
    int n = lane + 32 * it;
    float acc = 0.f;
    #pragma unroll 8
    for (int c = 0; c < CDIM; ++c) acc += arow[w * CDIM + c] * fb[(size_t)c * NN + n];
    float d = 2.f * acc - sqm - sq[b * NN + n];
    bool better = (d > tv[KNNK - 1]) ||
                  (d == tv[KNNK - 1] && n < ti[KNNK - 1]);
    if (better) {
      tv[KNNK - 1] = d; ti[KNNK - 1] = n;
      #pragma unroll
      for (int j = KNNK - 1; j > 0; --j) {      // one bubble pass keeps sorted
        float a = tv[j], bsw = tv[j - 1];
        int   ia = ti[j], ib = ti[j - 1];
        bool sw = (a > bsw) || (a == bsw && ia < ib);
        tv[j]     = sw ? bsw : a;  tv[j - 1] = sw ? a : bsw;
        ti[j]     = sw ? ib  : ia; ti[j - 1] = sw ? ia : ib;
      }
    }
  }

  // merge 32 sorted per-lane lists -> row top-16
  int p = 0;
  for (int r = 0; r < KNNK; ++r) {
    float bv = (p < KNNK) ? tv[p] : -3.402823466e38f;
    int   bi = (p < KNNK) ? ti[p] : 0x7FFFFFFF;
    int   bl = lane;
    #pragma unroll
    for (int off = 16; off > 0; off >>= 1) {
      float ov = __shfl_xor(bv, off, 32);
      int   oi = __shfl_xor(bi, off, 32);
      int   ol = __shfl_xor(bl, off, 32);
      bool take = (ov > bv) || (ov == bv && oi < bi);
      bv = take ? ov : bv; bi = take ? oi : bi; bl = take ? ol : bl;
    }
    if (lane == 0) idxo[((size_t)b * NN + m) * KNNK + r] = bi;
    if (lane == bl) ++p;
  }
}

// ---------------------------------------------------------------------------
// 3) per-node transforms via WMMA:
//    P[b,n,:] = W1[:, :128] * feats[b,:,n]
//    Q[b,n,:] = (W1[:,128:] - W1[:, :128]) * feats[b,:,n] + b1
//    GEMM with M = points (A = feats^T tile), N = HID cols, K = 128.
// ---------------------------------------------------------------------------
__global__ __launch_bounds__(256) void k_pq(const float* __restrict__ feats,
                                            const float* __restrict__ W1,
                                            const float* __restrict__ b1,
                                            float* __restrict__ P,
                                            float* __restrict__ Q) {
  int gw   = blockIdx.x * 8 + (threadIdx.x >> 5);   // global wave id
  int lane = threadIdx.x & 31;
  int ht = gw & 15;              // HID tile   (16 tiles of 16)
  int nt = (gw >> 4) & 127;      // point tile (128 tiles of 16)
  int b  = gw >> 11;
  int m0 = nt * 16;              // point base
  int c0 = ht * 16;              // hid-channel base
  int half = lane >> 4, l16 = lane & 15;

  const float* fb = feats + (size_t)b * CDIM * NN;
  v8f accP = {}; v8f accQ = {};

  for (int kc = 0; kc < CDIM; kc += 4) {
    int c = kc + 2 * half;
    v2f A;                                   // A[M=point, K=c]
    A.x = fb[(size_t)c * NN + m0 + l16];
    A.y = fb[(size_t)(c + 1) * NN + m0 + l16];
    int ch = c0 + l16;                       // B[K=c, N=ch]
    const float* wr = W1 + (size_t)ch * (2 * CDIM);
    float wk0 = wr[c],        wk1 = wr[c + 1];
    float wq0 = wr[CDIM + c] - wk0, wq1 = wr[CDIM + c + 1] - wk1;
    v2f Bk; Bk.x = wk0; Bk.y = wk1;
    v2f Bq; Bq.x = wq0; Bq.y = wq1;
    accP = wmma_f32(A, Bk, accP);
    accQ = wmma_f32(A, Bq, accQ);
  }

  float bias = b1[c0 + l16];
  size_t rowbase = (size_t)b * NN + m0 + 8 * half;
  #pragma unroll
  for (int j = 0; j < 8; ++j) {              // D: M = j + 8*half, N = l16
    size_t o = (rowbase + j) * HIDD + c0 + l16;
    P[o] = accP[j];
    Q[o] = accQ[j] + bias;
  }
}

// ---------------------------------------------------------------------------
// 4) second layer, fused gather + relu + GEMM + max over k:
//    e[b,o,n] = max_k ( W2 * relu(P[idx[b,n,k]] + Q[n]) )[o] + b2[o]
//    One WG per (b, 16-point tile). H tile (256x16) staged in LDS per k.
// ---------------------------------------------------------------------------
template <int OUTD>
__global__ __launch_bounds__(256) void k_conv2(const float* __restrict__ P,
                                               const float* __restrict__ Q,
                                               const int*   __restrict__ idx,
                                               const float* __restrict__ W2,
                                               const float* __restrict__ b2,
                                               float* __restrict__ e) {
  constexpr int OT   = (OUTD + 15) / 16;   // out tiles
  constexpr int MAXT = (OT + 7) / 8;       // tiles per wave (8 waves)
  __shared__ float H[16 * HPITCH];

  int b  = blockIdx.x >> 7;
  int n0 = (blockIdx.x & 127) * 16;
  int w = threadIdx.x >> 5, lane = threadIdx.x & 31;
  int half = lane >> 4, l16 = lane & 15;

  const float* Pb = P + (size_t)b * NN * HIDD;
  const float* Qb = Q + (size_t)b * NN * HIDD;

  v8f amax[MAXT];
  #pragma unroll
  for (int t = 0; t < MAXT; ++t)
    #pragma unroll
    for (int i = 0; i < 8; ++i) amax[t][i] = -3.402823466e38f;

  for (int k = 0; k < KNNK; ++k) {
    __syncthreads();                         // previous H fully consumed
    #pragma unroll
    for (int jj = 0; jj < 2; ++jj) {         // wave w builds points 2w, 2w+1
      int j  = 2 * w + jj;
      int gi = idx[((size_t)b * NN + n0 + j) * KNNK + k];
      const float* Pc = Pb + (size_t)gi * HIDD;
      const float* Qc = Qb + (size_t)(n0 + j) * HIDD;
      #pragma unroll
      for (int i = 0; i < HIDD / 32; ++i) {
        int c = lane + 32 * i;
        float v = Pc[c] + Qc[c];
        H[j * HPITCH + c] = v > 0.f ? v : 0.f;
      }
    }
    __syncthreads();

    #pragma unroll
    for (int t = 0; t < MAXT; ++t) {
      int ot = w + 8 * t;
      if (ot < OT) {
        v8f acc = {};
        for (int kc = 0; kc < HIDD; kc += 4) {
          int c = kc + 2 * half;
          int o = ot * 16 + l16;             // A[M=o, K=c] from W2
          v2f A;
          A.x = (o < OUTD) ? W2[(size_t)o * HIDD + c]     : 0.f;
          A.y = (o < OUTD) ? W2[(size_t)o * HIDD + c + 1] : 0.f;
          v2f Bv;                            // B[K=c, N=point l16] from LDS
          Bv.x = H[l16 * HPITCH + c];
          Bv.y = H[l16 * HPITCH + c + 1];
          acc = wmma_f32(A, Bv, acc);
        }
        #pragma unroll
        for (int i = 0; i < 8; ++i)
          amax[t][i] = fmaxf(amax[t][i], acc[i]);
      }
    }
  }

  #pragma unroll
  for (int t = 0; t < MAXT; ++t) {
    int ot = w + 8 * t;
    if (ot < OT) {
      #pragma unroll
      for (int j = 0; j < 8; ++j) {
        int o = ot * 16 + j + 8 * half;
        int n = n0 + l16;
        if (o < OUTD)
          e[((size_t)b * OUTD + o) * NN + n] = amax[t][j] + b2[o];
      }
    }
  }
}

// ---------------------------------------------------------------------------
// 5) direction head: softmax over 43 hypotheses, einsum with hyp, normalize,
//    scale, add to xyz. One thread per (b,n); float4 (over u) stores.
// ---------------------------------------------------------------------------
__global__ __launch_bounds__(256) void k_xyz(const float* __restrict__ xyzs,
                                             const float* __restrict__ ew,
                                             const float* __restrict__ es,
                                             const float* __restrict__ hyp,
                                             float* __restrict__ out) {
  int t = blockIdx.x * blockDim.x + threadIdx.x;
  if (t >= BB * NN) return;
  int b = t / NN, n = t % NN;
  const float* ewb = ew + (size_t)b * OUTW * NN + n;

  float r0[UPR], r1[UPR], r2[UPR];
  for (int u = 0; u < UPR; ++u) {
    float wv[NHYP], mx = -3.402823466e38f;
    #pragma unroll
    for (int h = 0; h < NHYP; ++h) {
      wv[h] = ewb[(size_t)(h * UPR + u) * NN];
      mx = fmaxf(mx, wv[h]);
    }
    float s = 0.f;
    #pragma unroll
    for (int h = 0; h < NHYP; ++h) { wv[h] = __expf(wv[h] - mx); s += wv[h]; }
    float inv = 1.f / s;
    float d0 = 0.f, d1 = 0.f, d2 = 0.f;
    #pragma unroll
    for (int h = 0; h < NHYP; ++h) {
      float wh = wv[h] * inv;
      d0 += wh * hyp[h * 3 + 0];
      d1 += wh * hyp[h * 3 + 1];
      d2 += wh * hyp[h * 3 + 2];
    }
    float nrm = fmaxf(sqrtf(d0 * d0 + d1 * d1 + d2 * d2), 1e-12f);
    float sc  = es[((size_t)b * OUTS + u) * NN + n] / nrm;
    r0[u] = d0 * sc; r1[u] = d1 * sc; r2[u] = d2 * sc;
  }
  float4* o4 = (float4*)out;
  float x0 = xyzs[((size_t)b * 3 + 0) * NN + n];
  float x1 = xyzs[((size_t)b * 3 + 1) * NN + n];
  float x2 = xyzs[((size_t)b * 3 + 2) * NN + n];
  o4[((size_t)b * 3 + 0) * NN + n] = make_float4(x0 + r0[0], x0 + r0[1], x0 + r0[2], x0 + r0[3]);
  o4[((size_t)b * 3 + 1) * NN + n] = make_float4(x1 + r1[0], x1 + r1[1], x1 + r1[2], x1 + r1[3]);
  o4[((size_t)b * 3 + 2) * NN + n] = make_float4(x2 + r2[0], x2 + r2[1], x2 + r2[2], x2 + r2[3]);
}

// ---------------------------------------------------------------------------
// 6) feature head: out_feats[b,c,n,u] = e_f[b, c*4+u, n] + feats[b,c,n]
// ---------------------------------------------------------------------------
__global__ __launch_bounds__(256) void k_feat(const float* __restrict__ feats,
                                              const float* __restrict__ ef,
                                              float* __restrict__ out) {
  int t = blockIdx.x * blockDim.x + threadIdx.x;
  if (t >= BB * CDIM * NN) return;
  int n = t % NN, c = (t / NN) % CDIM, b = t / (NN * CDIM);
  float f = feats[((size_t)b * CDIM + c) * NN + n];
  const float* efb = ef + ((size_t)b * OUTF + c * UPR) * NN + n;
  float4 o = make_float4(efb[0] + f, efb[(size_t)NN] + f,
                         efb[(size_t)2 * NN] + f, efb[(size_t)3 * NN] + f);
  ((float4*)out)[((size_t)b * CDIM + c) * NN + n] = o;
}

// ---------------------------------------------------------------------------
extern "C" void kernel_launch(void* const* d_in, const int* in_sizes, int n_in,
                              void* d_out, int out_size, void* d_ws, size_t ws_size,
                              hipStream_t stream) {
  (void)in_sizes; (void)n_in; (void)out_size; (void)ws_size;
  const float* xyzs  = (const float*)d_in[0];
  const float* feats = (const float*)d_in[1];
  const float* hyp   = (const float*)d_in[2];
  const float* Ww1 = (const float*)d_in[3];  const float* bw1 = (const float*)d_in[4];
  const float* Ww2 = (const float*)d_in[5];  const float* bw2 = (const float*)d_in[6];
  const float* Ws1 = (const float*)d_in[7];  const float* bs1 = (const float*)d_in[8];
  const float* Ws2 = (const float*)d_in[9];  const float* bs2 = (const float*)d_in[10];
  const float* Wf1 = (const float*)d_in[11]; const float* bf1 = (const float*)d_in[12];
  const float* Wf2 = (const float*)d_in[13]; const float* bf2 = (const float*)d_in[14];

  // workspace carve-up (256B aligned)
  char* ws = (char*)d_ws;
  size_t off = 0;
  auto carve = [&](size_t bytes) { char* p = ws + off; off = (off + bytes + 255) & ~(size_t)255; return p; };
  float* sq  = (float*)carve((size_t)BB * NN * 4);
  int*   idx = (int*)  carve((size_t)BB * NN * KNNK * 4);
  float* P   = (float*)carve((size_t)BB * NN * HIDD * 4);
  float* Q   = (float*)carve((size_t)BB * NN * HIDD * 4);
  float* ew  = (float*)carve((size_t)BB * OUTW * NN * 4);
  float* es  = (float*)carve((size_t)BB * OUTS * NN * 4);
  float* ef  = (float*)carve((size_t)BB * OUTF * NN * 4);

  float* out_xyz  = (float*)d_out;
  float* out_feat = (float*)d_out + (size_t)BB * 3 * NN * UPR;

  // 1) norms + 2) KNN (shared by all three edge convs)
  k_sq  <<<(BB * NN + 255) / 256, 256, 0, stream>>>(feats, sq);
  k_topk<<<BB * NN / 8,           256, 0, stream>>>(feats, sq, idx);

  // 3/4) three edge convs; P/Q reused (stream serializes kernels)
  const int PQ_BLOCKS = BB * (NN / 16) * (HIDD / 16) / 8;  // 2048
  const int C2_BLOCKS = BB * (NN / 16);                    // 1024
  k_pq<<<PQ_BLOCKS, 256, 0, stream>>>(feats, Ww1, bw1, P, Q);
  k_conv2<OUTW><<<C2_BLOCKS, 256, 0, stream>>>(P, Q, idx, Ww2, bw2, ew);
  k_pq<<<PQ_BLOCKS, 256, 0, stream>>>(feats, Ws1, bs1, P, Q);
  k_conv2<OUTS><<<C2_BLOCKS, 256, 0, stream>>>(P, Q, idx, Ws2, bs2, es);
  k_pq<<<PQ_BLOCKS, 256, 0, stream>>>(feats, Wf1, bf1, P, Q);
  k_conv2<OUTF><<<C2_BLOCKS, 256, 0, stream>>>(P, Q, idx, Wf2, bf2, ef);

  // 5/6) heads
  k_xyz <<<(BB * NN + 255) / 256,        256, 0, stream>>>(xyzs, ew, es, hyp, out_xyz);
  k_feat<<<(BB * CDIM * NN + 255) / 256, 256, 0, stream>>>(feats, ef, out_feat);
}